// PagedAttention_63943473103532
// MI455X (gfx1250) — compile-verified
//
#include <hip/hip_runtime.h>
#include <cstdint>

// ---- problem constants (match reference) ----
#define BQ     16
#define HQ     32
#define KVHQ   8
#define GQ     4          // H / KVH
#define DQ     128
#define BLK    16
#define NFETCH 256        // CACHE_LEN / BLOCK
#define NTILES 257        // cache tiles + 1 new-token tile
#define NWAVES 8
#define SCALE  0.08838834764831845f   // 1/sqrt(128)

#define TILE_F      (BLK * DQ)                   // 2048 floats (8KB)
// per-wave: double-buffered K/V tiles + 16x16 P buffer
#define WAVE_SMEM_F (4 * TILE_F + BLK * BLK)     // 8448 floats
#define SMEM_FLOATS (NWAVES * WAVE_SMEM_F + 16 * DQ + 2 * DQ)
#define SMEM_BYTES  (SMEM_FLOATS * 4)            // ~273 KB < 320 KB WGP LDS

typedef __attribute__((ext_vector_type(2))) float v2f;
typedef __attribute__((ext_vector_type(8))) float v8f;

// ---- CDNA5 WMMA: D = A(16x4 f32) * B(4x16 f32) + C(16x16 f32) ----
__device__ __forceinline__ v8f wmma4(v2f a, v2f b, v8f c) {
  return __builtin_amdgcn_wmma_f32_16x16x4_f32(
      /*neg_a=*/false, a, /*neg_b=*/false, b,
      /*c_mod=*/(short)0, c, /*reuse_a=*/false, /*reuse_b=*/false);
}

// ---- async global->LDS (CDNA5) ----
#if __has_builtin(__builtin_amdgcn_global_load_async_to_lds_b128)
#define HAVE_ASYNC_LDS 1
typedef __attribute__((__vector_size__(4 * sizeof(int)))) int b128_t;
typedef __attribute__((address_space(1))) b128_t* gp128_t;
typedef __attribute__((address_space(3))) b128_t* lp128_t;
__device__ __forceinline__ void async_b128(const float* gsrc, float* ldst) {
  __builtin_amdgcn_global_load_async_to_lds_b128(
      (gp128_t)(uintptr_t)gsrc, (lp128_t)(uint32_t)(uintptr_t)ldst, 0, 0);
}
#endif

// wait until at most N async-LDS loads remain outstanding
__device__ __forceinline__ void wait_async0() {
#if __has_builtin(__builtin_amdgcn_s_wait_asynccnt)
  __builtin_amdgcn_s_wait_asynccnt(0);
  asm volatile("" ::: "memory");
#else
  asm volatile("s_wait_asynccnt 0x0" ::: "memory");
#endif
}
__device__ __forceinline__ void wait_async32() {
#if __has_builtin(__builtin_amdgcn_s_wait_asynccnt)
  __builtin_amdgcn_s_wait_asynccnt(32);
  asm volatile("" ::: "memory");
#else
  asm volatile("s_wait_asynccnt 0x20" ::: "memory");
#endif
}

// stage one 16x128 K tile + V tile into the given LDS buffers
__device__ __forceinline__ void stage_tile(int t, int b, int kvh, int lane,
                                           const float* __restrict__ Kc,
                                           const float* __restrict__ Vc,
                                           const int* __restrict__ fetch,
                                           const float* sKr, const float* sVn,
                                           float* kt, float* vt) {
  if (t < NFETCH) {
    const long long base =
        (long long)fetch[b * NFETCH + t] * (KVHQ * DQ) + kvh * (BLK * DQ);
    const float* gk = Kc + base;
    const float* gv = Vc + base;
#ifdef HAVE_ASYNC_LDS
#pragma unroll
    for (int c = 0; c < 16; ++c) {
      async_b128(gk + (c * 32 + lane) * 4, kt + (c * 32 + lane) * 4);
      async_b128(gv + (c * 32 + lane) * 4, vt + (c * 32 + lane) * 4);
    }
#else
    const float4* gk4 = (const float4*)gk;
    const float4* gv4 = (const float4*)gv;
#pragma unroll
    for (int c = 0; c < 16; ++c) {
      ((float4*)kt)[c * 32 + lane] = gk4[c * 32 + lane];
      ((float4*)vt)[c * 32 + lane] = gv4[c * 32 + lane];
    }
#endif
  } else {
    // new-token tile: row 0 = rope'd K / new V, rows 1..15 zero (masked later)
    for (int i = lane; i < TILE_F; i += 32) {
      const int row = i >> 7, d = i & 127;
      kt[i] = (row == 0) ? sKr[d] : 0.0f;
      vt[i] = (row == 0) ? sVn[d] : 0.0f;
    }
  }
}

__global__ __launch_bounds__(256) void paged_attn_decode(
    const float* __restrict__ Q, const float* __restrict__ Knew,
    const float* __restrict__ Vnew, const float* __restrict__ Kc,
    const float* __restrict__ Vc, const float* __restrict__ cosv,
    const float* __restrict__ sinv, const int* __restrict__ fetch,
    float* __restrict__ out) {
  extern __shared__ float smem[];

  const int b    = blockIdx.x / KVHQ;
  const int kvh  = blockIdx.x % KVHQ;
  const int tid  = threadIdx.x;
  const int wave = tid >> 5;
  const int lane = tid & 31;
  const int qn   = lane & 15;
  const int half = lane >> 4;

  float* wbase  = &smem[wave * WAVE_SMEM_F];
  float* kt_cur = wbase;                    // buffer 0: K
  float* vt_cur = wbase + TILE_F;           // buffer 0: V
  float* kt_nxt = wbase + 2 * TILE_F;       // buffer 1: K
  float* vt_nxt = wbase + 3 * TILE_F;       // buffer 1: V
  float* pb     = wbase + 4 * TILE_F;       // 16x16 P buffer
  float* sQ  = &smem[NWAVES * WAVE_SMEM_F]; // 16x128 rope'd Q (rows 4..15 = 0)
  float* sKr = sQ + 16 * DQ;
  float* sVn = sKr + DQ;

  // ---- RoPE for Q (4 heads of this group) and new K; stage new V ----
  if (tid < DQ) {
    const int d = tid;
    const float c = cosv[b * DQ + d];
    const float s = sinv[b * DQ + d];
    const float* kp = Knew + (b * KVHQ + kvh) * DQ;
    const float kro = (d < 64) ? -kp[d + 64] : kp[d - 64];
    sKr[d] = kp[d] * c + kro * s;
    sVn[d] = Vnew[(b * KVHQ + kvh) * DQ + d];
#pragma unroll
    for (int g = 0; g < GQ; ++g) {
      const float* qp = Q + (b * HQ + g * KVHQ + kvh) * DQ;
      const float qro = (d < 64) ? -qp[d + 64] : qp[d - 64];
      sQ[g * DQ + d] = qp[d] * c + qro * s;
    }
  }
  for (int i = tid; i < (16 - GQ) * DQ; i += 256) sQ[GQ * DQ + i] = 0.0f;
  __syncthreads();

  // ---- resident Q^T B-fragments ----
  v2f qb[32];
#pragma unroll
  for (int j = 0; j < 32; ++j)
    qb[j] = *(const v2f*)&sQ[qn * DQ + 4 * j + 2 * half];

  v8f o[8];
#pragma unroll
  for (int dt = 0; dt < 8; ++dt)
    o[dt] = (v8f){0.f, 0.f, 0.f, 0.f, 0.f, 0.f, 0.f, 0.f};
  float m_run = -INFINITY, l_run = 0.0f;

  // prime the pipeline: stage first tile into the current buffer
  stage_tile(wave, b, kvh, lane, Kc, Vc, fetch, sKr, sVn, kt_cur, vt_cur);

  for (int t = wave; t < NTILES; t += NWAVES) {
    // ---- prefetch next tile into the alternate buffer ----
    const int tn = t + NWAVES;
    const bool pre_async = (tn < NFETCH);     // prefetch issues 32 asyncs
    if (tn < NTILES)
      stage_tile(tn, b, kvh, lane, Kc, Vc, fetch, sKr, sVn, kt_nxt, vt_nxt);
    // current buffer ready when <=32 (the prefetch batch) remain outstanding
    if (pre_async) wait_async32(); else wait_async0();

    float* kt = kt_cur;
    float* vt = vt_cur;

    // ---- S^T = K(16x128) x Q^T(128x16): M=key, N=query ----
    v8f s = (v8f){0.f, 0.f, 0.f, 0.f, 0.f, 0.f, 0.f, 0.f};
#pragma unroll
    for (int j = 0; j < 32; ++j) {
      const v2f a = *(const v2f*)&kt[qn * DQ + 4 * j + 2 * half];
      s = wmma4(a, qb[j], s);
    }

    // ---- online softmax: row = query = (lane&15) ----
    const bool last = (t == NFETCH);
    float sv[8];
    float rmax = -INFINITY;
#pragma unroll
    for (int r = 0; r < 8; ++r) {
      float x = s[r] * SCALE;                            // (key=r+8h, qn)
      if (last && (r != 0 || half != 0)) x = -INFINITY;  // only key 0 valid
      sv[r] = x;
      rmax = fmaxf(rmax, x);
    }
    rmax = fmaxf(rmax, __shfl_xor(rmax, 16, 32));
    const float m_new = fmaxf(m_run, rmax);
    float p[8];
    float rsum = 0.0f;
#pragma unroll
    for (int r = 0; r < 8; ++r) {
      p[r] = __expf(sv[r] - m_new);
      rsum += p[r];
    }
    rsum += __shfl_xor(rsum, 16, 32);
    const float corr = __expf(m_run - m_new);
    l_run = l_run * corr + rsum;
    m_run = m_new;

    // rescale O accumulators: row r+8h needs corr held at lane (r+8h)
#pragma unroll
    for (int r = 0; r < 8; ++r) {
      const float cr = __shfl(corr, r + 8 * half, 32);
#pragma unroll
      for (int dt = 0; dt < 8; ++dt) o[dt][r] *= cr;
    }

    // ---- re-fragment P through LDS: pb[query][key] ----
    *(float4*)&pb[qn * 16 + 8 * half] = make_float4(p[0], p[1], p[2], p[3]);
    *(float4*)&pb[qn * 16 + 8 * half + 4] = make_float4(p[4], p[5], p[6], p[7]);

    v2f pa[4];
#pragma unroll
    for (int j = 0; j < 4; ++j)
      pa[j] = *(const v2f*)&pb[qn * 16 + 4 * j + 2 * half];

    // ---- O(16x128) += P(16x16) x V(16x128) ----
#pragma unroll
    for (int dt = 0; dt < 8; ++dt) {
#pragma unroll
      for (int j = 0; j < 4; ++j) {
        const int k0 = 4 * j + 2 * half;
        v2f bv;
        bv.x = vt[k0 * DQ + dt * 16 + qn];
        bv.y = vt[(k0 + 1) * DQ + dt * 16 + qn];
        o[dt] = wmma4(pa[j], bv, o[dt]);
      }
    }

    // swap double buffers
    float* tk = kt_cur; kt_cur = kt_nxt; kt_nxt = tk;
    float* tv = vt_cur; vt_cur = vt_nxt; vt_nxt = tv;
  }

  // ---- cross-wave combine (reuse wave-0 LDS region after barrier) ----
  __syncthreads();
  float* obuf = smem;                       // [8][4][128]
  float* mbuf = smem + NWAVES * GQ * DQ;    // [8][4]
  float* lbuf = mbuf + NWAVES * GQ;         // [8][4]

  if (half == 0) {
#pragma unroll
    for (int g = 0; g < GQ; ++g)
#pragma unroll
      for (int dt = 0; dt < 8; ++dt)
        obuf[(wave * GQ + g) * DQ + dt * 16 + qn] = o[dt][g];
    if (qn < GQ) {
      mbuf[wave * GQ + qn] = m_run;
      lbuf[wave * GQ + qn] = l_run;
    }
  }
  __syncthreads();

  for (int e = tid; e < GQ * DQ; e += 256) {
    const int g = e >> 7, d = e & 127;
    float M = -INFINITY;
    for (int w = 0; w < NWAVES; ++w) M = fmaxf(M, mbuf[w * GQ + g]);
    float L = 0.0f, acc = 0.0f;
    for (int w = 0; w < NWAVES; ++w) {
      const float wgt = __expf(mbuf[w * GQ + g] - M);
      L += wgt * lbuf[w * GQ + g];
      acc += wgt * obuf[(w * GQ + g) * DQ + d];
    }
    out[(b * HQ + g * KVHQ + kvh) * DQ + d] = acc / L;
  }
}

extern "C" void kernel_launch(void* const* d_in, const int* in_sizes, int n_in,
                              void* d_out, int out_size, void* d_ws,
                              size_t ws_size, hipStream_t stream) {
  const float* Q    = (const float*)d_in[0];
  const float* K    = (const float*)d_in[1];
  const float* V    = (const float*)d_in[2];
  const float* Kc   = (const float*)d_in[3];
  const float* Vc   = (const float*)d_in[4];
  const float* cosv = (const float*)d_in[5];
  const float* sinv = (const float*)d_in[6];
  const int* fetch  = (const int*)d_in[7];
  // d_in[8] = save_slots: unused by the reference computation
  float* out = (float*)d_out;

  hipLaunchKernelGGL(paged_attn_decode, dim3(BQ * KVHQ), dim3(256), SMEM_BYTES,
                     stream, Q, K, V, Kc, Vc, cosv, sinv, fetch, out);
}